// HRLModel_28286654612137
// MI455X (gfx1250) — compile-verified
//
#include <hip/hip_runtime.h>
#include <hip/hip_bf16.h>

// Problem constants (from reference): V=32000, D=512, L=4096, S=256
#define Dc 512
#define Lc 4096
#define Sc 256

typedef __attribute__((ext_vector_type(2))) float v2f;
typedef __attribute__((ext_vector_type(8))) float v8f;

// ---------------------------------------------------------------------------
// Deterministic 256-thread block reduction (fixed tree, no atomics)
// ---------------------------------------------------------------------------
__device__ __forceinline__ float block_reduce_sum256(float v, float* red) {
    const int t = threadIdx.x;
    red[t] = v;
    __syncthreads();
    #pragma unroll
    for (int off = 128; off > 0; off >>= 1) {
        if (t < off) red[t] += red[t + off];
        __syncthreads();
    }
    float r = red[0];
    __syncthreads();
    return r;
}

// ---------------------------------------------------------------------------
// Kernel 0: pack the zero-padded B panel, transposed so each lane's fragment
// is a contiguous 8-byte load: wpadT[n*Dc + k] = W[k][n], cols 6..15 = 0.
// ---------------------------------------------------------------------------
__global__ void hrl_pack_w(const float* __restrict__ w1,
                           const float* __restrict__ w2,
                           const float* __restrict__ w3,
                           float* __restrict__ wpadT) {
    const int idx = blockIdx.x * 256 + threadIdx.x;   // 0 .. 16*Dc-1
    const int n = idx >> 9;      // column 0..15
    const int k = idx & (Dc - 1);
    float v = 0.f;
    if (n < 6) {
        const float* w = (n < 2) ? w1 : (n < 4) ? w2 : w3;
        v = w[k * 2 + (n & 1)];
    }
    wpadT[n * Dc + k] = v;
}

// ---------------------------------------------------------------------------
// Kernel 1: z[L][6] = embd[x] @ [w1|w2|w3]  via V_WMMA_F32_16X16X4_F32
// One wave (32 threads) per 16-row tile; K loops over D in steps of 4.
// A layout (32-bit A 16x4): lane m=lane&15; VGPR0/1 hold K = koff, koff+1
//   with koff = 0 (lanes 0-15) or 2 (lanes 16-31).
// B layout (mirror): lane n=lane&15; VGPR0/1 hold K rows koff, koff+1.
// C layout: VGPR r -> M = r + (lane>=16 ? 8 : 0), N = lane&15.
// Branch-free inner loop: A and B fragments are both single b64 loads.
// ---------------------------------------------------------------------------
__global__ void hrl_gemm_wmma(const int* __restrict__ x,
                              const float* __restrict__ embd,
                              const float* __restrict__ wpadT,
                              float* __restrict__ z) {
    const int lane = threadIdx.x;            // 0..31, one wave per block
    const int tile = blockIdx.x;             // 16 rows per tile
    const int m    = lane & 15;
    const int koff = (lane >> 4) << 1;       // 0 or 2
    const int row  = tile * 16 + m;
    const int n    = lane & 15;              // B / C column this lane owns

    const float* acol = embd + (long long)x[row] * Dc + koff;
    const float* bcol = wpadT + n * Dc + koff;

    v8f c = {0.f, 0.f, 0.f, 0.f, 0.f, 0.f, 0.f, 0.f};

    #pragma unroll 4
    for (int k0 = 0; k0 < Dc; k0 += 4) {
        const v2f a = *(const v2f*)(acol + k0);   // 8B aligned (koff even)
        const v2f b = *(const v2f*)(bcol + k0);
        // 8 args: (neg_a, A, neg_b, B, c_mod, C, reuse_a, reuse_b)
        c = __builtin_amdgcn_wmma_f32_16x16x4_f32(
                false, a, false, b, (short)0, c, false, false);
    }

    if (n < 6) {
        const int mhi = (lane >> 4) << 3;    // +8 for lanes 16..31
        #pragma unroll
        for (int r = 0; r < 8; ++r) {
            const int M = r + mhi;
            z[(tile * 16 + M) * 6 + n] = c[r];
        }
    }
}

// ---------------------------------------------------------------------------
// Kernel 2: per-position 2-class log-softmax for all three heads + entropy
// partial sums (16 blocks x 256 threads covers L exactly).
// ---------------------------------------------------------------------------
__device__ __forceinline__ void lsm2(float l0, float l1,
                                     float& lp0, float& lp1, float& nent) {
    const float mx = fmaxf(l0, l1);
    const float e0 = __expf(l0 - mx);
    const float e1 = __expf(l1 - mx);
    const float s  = e0 + e1;
    const float lse = mx + __logf(s);
    lp0 = l0 - lse;
    lp1 = l1 - lse;
    // -(p0*lp0 + p1*lp1) / ln(2)
    nent = -((e0 * lp0 + e1 * lp1) / s) * 1.4426950408889634f;
}

__global__ void hrl_head_stats(const float* __restrict__ z,
                               const float* __restrict__ bias1,
                               const float* __restrict__ bias2,
                               const float* __restrict__ bias3,
                               float* __restrict__ lp1t,
                               float* __restrict__ lp2t,
                               float* __restrict__ lp3t,
                               float* __restrict__ nentP /* [3*16] */) {
    __shared__ float red[256];
    const int j = blockIdx.x * 256 + threadIdx.x;   // j < L always

    float n1 = 0.f, n2 = 0.f, n3 = 0.f;

    // head 1 : logits = z[j][0..1] + b1
    {
        float lp0, lp1, ne;
        lsm2(z[j * 6 + 0] + bias1[0], z[j * 6 + 1] + bias1[1], lp0, lp1, ne);
        lp1t[2 * j]     = lp0;
        lp1t[2 * j + 1] = lp1;
        n1 = ne;
    }
    // head 2 : logits = (z[j][2..3] + z[j+1][2..3]) / 2 + b2
    if (j < Lc - 1) {
        float lp0, lp1, ne;
        const float l0 = (z[j * 6 + 2] + z[(j + 1) * 6 + 2]) * 0.5f + bias2[0];
        const float l1 = (z[j * 6 + 3] + z[(j + 1) * 6 + 3]) * 0.5f + bias2[1];
        lsm2(l0, l1, lp0, lp1, ne);
        lp2t[2 * j]     = lp0;
        lp2t[2 * j + 1] = lp1;
        n2 = ne;
    }
    // head 3 : logits = (z[j] + z[j+1] + z[j+2])[4..5] / 3 + b3
    if (j < Lc - 2) {
        float lp0, lp1, ne;
        const float l0 = (z[j * 6 + 4] + z[(j + 1) * 6 + 4] + z[(j + 2) * 6 + 4]) * (1.f / 3.f) + bias3[0];
        const float l1 = (z[j * 6 + 5] + z[(j + 1) * 6 + 5] + z[(j + 2) * 6 + 5]) * (1.f / 3.f) + bias3[1];
        lsm2(l0, l1, lp0, lp1, ne);
        lp3t[2 * j]     = lp0;
        lp3t[2 * j + 1] = lp1;
        n3 = ne;
    }

    const float s1 = block_reduce_sum256(n1, red);
    const float s2 = block_reduce_sum256(n2, red);
    const float s3 = block_reduce_sum256(n3, red);
    if (threadIdx.x == 0) {
        nentP[blockIdx.x]      = s1;
        nentP[16 + blockIdx.x] = s2;
        nentP[32 + blockIdx.x] = s3;
    }
}

// ---------------------------------------------------------------------------
// Kernel 3: per (sample, position) gumbel argmax -> raw booleans written as
// float 0/1 straight into the d_out slots, plus nll1 per-sample reduction.
// One block per sample; threads stride positions. Deterministic tree reduce.
// ---------------------------------------------------------------------------
__global__ void hrl_sample_select(const float* __restrict__ lp1t,
                                  const float* __restrict__ lp2t,
                                  const float* __restrict__ lp3t,
                                  const float* __restrict__ g1,
                                  const float* __restrict__ g2,
                                  const float* __restrict__ g3,
                                  const unsigned char* __restrict__ al1,
                                  const unsigned char* __restrict__ al2,
                                  const unsigned char* __restrict__ al3,
                                  float* __restrict__ ob1,
                                  float* __restrict__ ob2,
                                  float* __restrict__ ob3,
                                  float* __restrict__ nll1w) {
    __shared__ float red[256];
    const int s = blockIdx.x;
    float acc = 0.f;

    for (int j = threadIdx.x; j < Lc; j += 256) {
        // head 1 (also drives nll1, independent of align mask / conflicts)
        {
            const float lp0 = lp1t[2 * j], lp1 = lp1t[2 * j + 1];
            const float* g = g1 + ((long long)s * Lc + j) * 2;
            const bool sel = (lp0 + g[0]) >= (lp1 + g[1]);   // argmax, ties->0
            acc += sel ? -lp0 : -lp1;
            ob1[(long long)s * Lc + j] = (sel && al1[j]) ? 1.f : 0.f;
        }
        if (j < Lc - 1) {
            const float lp0 = lp2t[2 * j], lp1 = lp2t[2 * j + 1];
            const float* g = g2 + ((long long)s * (Lc - 1) + j) * 2;
            const bool sel = (lp0 + g[0]) >= (lp1 + g[1]);
            ob2[(long long)s * (Lc - 1) + j] = (sel && al2[j]) ? 1.f : 0.f;
        }
        if (j < Lc - 2) {
            const float lp0 = lp3t[2 * j], lp1 = lp3t[2 * j + 1];
            const float* g = g3 + ((long long)s * (Lc - 2) + j) * 2;
            const bool sel = (lp0 + g[0]) >= (lp1 + g[1]);
            ob3[(long long)s * (Lc - 2) + j] = (sel && al3[j]) ? 1.f : 0.f;
        }
    }

    const float tot = block_reduce_sum256(acc, red);
    if (threadIdx.x == 0) nll1w[s] = tot;
}

// ---------------------------------------------------------------------------
// Kernel 4: sequential conflict resolution (one thread per sample), nll2/nll3
// accumulation, and final nent / nll outputs. 1 block x 256 threads.
// ---------------------------------------------------------------------------
__global__ void hrl_finalize(const float* __restrict__ lp2t,
                             const float* __restrict__ lp3t,
                             const float* __restrict__ nentP,
                             const float* __restrict__ nll1w,
                             float* __restrict__ out) {
    const int s = threadIdx.x;   // one sample per thread

    float* ob1 = out + 2 * Sc;
    float* ob2 = ob1 + (long long)Sc * Lc;
    float* ob3 = ob2 + (long long)Sc * (Lc - 1);
    float* r1 = ob1 + (long long)s * Lc;
    float* r2 = ob2 + (long long)s * (Lc - 1);
    float* r3 = ob3 + (long long)s * (Lc - 2);

    // ---- b3: span-3 self conflict (pass1: dep i-1 on pass1; pass2: dep i-2
    // on pass2; pass2 outputs at 0,1 equal pass1 outputs) + nll3
    float nll3 = 0.f;
    int c1p = 0;            // pass-1 value at i-1
    int c2p1 = 0, c2p2 = 0; // pass-2 values at i-1, i-2
    for (int i = 0; i < Lc - 2; ++i) {
        const int raw = (r3[i] != 0.f);
        const int c1  = raw & (c1p ^ 1);
        const int c2  = (i < 2) ? c1 : (c1 & (c2p2 ^ 1));
        r3[i] = (float)c2;
        nll3 += c2 ? -lp3t[2 * i] : -lp3t[2 * i + 1];
        c1p = c1;
        c2p2 = c2p1;
        c2p1 = c2;
    }

    // ---- b2: cleared by b3 window {j-2, j-1, j, j+1}, then span-2 scan + nll2
    float nll2 = 0.f;
    int prev = 0;
    for (int j = 0; j < Lc - 1; ++j) {
        int hit = 0;
        if (j < Lc - 2)     hit |= (r3[j]     != 0.f);
        if (j >= 1)         hit |= (r3[j - 1] != 0.f);
        if (j + 1 < Lc - 2) hit |= (r3[j + 1] != 0.f);
        if (j >= 2)         hit |= (r3[j - 2] != 0.f);
        const int raw = (r2[j] != 0.f) & (hit ^ 1);
        const int b   = raw & (prev ^ 1);
        r2[j] = (float)b;
        nll2 += b ? -lp2t[2 * j] : -lp2t[2 * j + 1];
        prev = b;
    }

    // ---- b1: cleared by b3 window {j, j-1, j-2} and b2 window {j, j-1}
    for (int j = 0; j < Lc; ++j) {
        int hit = 0;
        if (j < Lc - 2)                hit |= (r3[j]     != 0.f);
        if (j >= 1 && j - 1 < Lc - 2)  hit |= (r3[j - 1] != 0.f);
        if (j >= 2 && j - 2 < Lc - 2)  hit |= (r3[j - 2] != 0.f);
        if (j < Lc - 1)                hit |= (r2[j]     != 0.f);
        if (j >= 1)                    hit |= (r2[j - 1] != 0.f);
        const int b = (r1[j] != 0.f) & (hit ^ 1);
        r1[j] = (float)b;
    }

    // ---- scalar entropy (fixed-order sum of the 48 block partials)
    float sn1 = 0.f, sn2 = 0.f, sn3 = 0.f;
    for (int b = 0; b < 16; ++b) {
        sn1 += nentP[b];
        sn2 += nentP[16 + b];
        sn3 += nentP[32 + b];
    }
    const float nent = (sn1 / (float)Lc +
                        sn2 / (float)(Lc - 1) +
                        sn3 / (float)(Lc - 2)) * (1.f / 3.f);

    out[s]      = nent;
    out[Sc + s] = nll1w[s] + nll2 + nll3;
}

// ---------------------------------------------------------------------------
// Launch
// ---------------------------------------------------------------------------
extern "C" void kernel_launch(void* const* d_in, const int* in_sizes, int n_in,
                              void* d_out, int out_size, void* d_ws, size_t ws_size,
                              hipStream_t stream) {
    // setup_inputs order:
    // x, sample_num, align1, align2, align3, embd, w1, b1, w2, b2, w3, b3, g1, g2, g3
    const int*   x    = (const int*)d_in[0];
    const unsigned char* al1 = (const unsigned char*)d_in[2];  // jax bool = 1 byte
    const unsigned char* al2 = (const unsigned char*)d_in[3];
    const unsigned char* al3 = (const unsigned char*)d_in[4];
    const float* embd = (const float*)d_in[5];
    const float* w1   = (const float*)d_in[6];
    const float* b1   = (const float*)d_in[7];
    const float* w2   = (const float*)d_in[8];
    const float* b2   = (const float*)d_in[9];
    const float* w3   = (const float*)d_in[10];
    const float* b3   = (const float*)d_in[11];
    const float* g1   = (const float*)d_in[12];
    const float* g2   = (const float*)d_in[13];
    const float* g3   = (const float*)d_in[14];

    float* out = (float*)d_out;

    // workspace partition (~230 KB)
    float* wpadT = (float*)d_ws;        // 16 * Dc
    float* z     = wpadT + 16 * Dc;     // L*6
    float* lp1t  = z    + Lc * 6;       // L*2
    float* lp2t  = lp1t + Lc * 2;       // (L-1)*2, padded slot L*2
    float* lp3t  = lp2t + Lc * 2;       // (L-2)*2, padded slot L*2
    float* nentP = lp3t + Lc * 2;       // 48
    float* nll1w = nentP + 48;          // S

    float* ob1 = out + 2 * Sc;
    float* ob2 = ob1 + (long long)Sc * Lc;
    float* ob3 = ob2 + (long long)Sc * (Lc - 1);

    hipLaunchKernelGGL(hrl_pack_w, dim3((16 * Dc) / 256), dim3(256), 0, stream,
                       w1, w2, w3, wpadT);
    hipLaunchKernelGGL(hrl_gemm_wmma, dim3(Lc / 16), dim3(32), 0, stream,
                       x, embd, wpadT, z);
    hipLaunchKernelGGL(hrl_head_stats, dim3(Lc / 256), dim3(256), 0, stream,
                       z, b1, b2, b3, lp1t, lp2t, lp3t, nentP);
    hipLaunchKernelGGL(hrl_sample_select, dim3(Sc), dim3(256), 0, stream,
                       lp1t, lp2t, lp3t, g1, g2, g3, al1, al2, al3,
                       ob1, ob2, ob3, nll1w);
    hipLaunchKernelGGL(hrl_finalize, dim3(1), dim3(256), 0, stream,
                       lp2t, lp3t, nentP, nll1w, out);
}